// Flow_26774826123816
// MI455X (gfx1250) — compile-verified
//
#include <hip/hip_runtime.h>

// ---------------------------------------------------------------------------
// LTI Dopri5 scan on gfx1250, parallel-in-time, f32 WMMA (V_WMMA_F32_16X16X4_F32)
// ---------------------------------------------------------------------------
#define BATCH   256
#define T_LEN   4096
#define NS      32
#define NI      8
#define NO      8
#define NCHUNK  64
#define LCHUNK  (T_LEN / NCHUNK)   // 64 = 2^6 -> 6 repeated squarings for W^L
#define BT      (BATCH / 16)       // 16 batch tiles of 16 rows

// workspace layout (floats)
#define W_OFF   0                      // W   [32][32] row conv: x' = x @ W
#define W0_OFF  1024                   // W0  [8][32]
#define W1_OFF  1280                   // W1  [8][32]
#define WL_OFF  1536                   // W^LCHUNK [32][32]
#define TMP_OFF 2560                   // squaring scratch [32][32]
#define CT_OFF  3584                   // C^T padded [32][16]
#define DT_OFF  4096                   // D^T padded [8][16]
#define R_OFF   4224                   // residuals  [NCHUNK][BT][2][32][8]
#define RSZ     ((size_t)NCHUNK * BT * 2 * 32 * 8)
#define XST_OFF (R_OFF + RSZ)          // chunk start states, same shape

typedef float v2f __attribute__((ext_vector_type(2)));
typedef float v8f __attribute__((ext_vector_type(8)));

#define WMMA4(A_, B_, C_) \
  __builtin_amdgcn_wmma_f32_16x16x4_f32(false, (A_), false, (B_), (short)0, (C_), false, false)

// lane-halves swap within the wave: SWAPX16 (xor=0x10, and=0x1f) => offset 0x021F
static __device__ __forceinline__ float swz16(float x) {
  return __int_as_float(__builtin_amdgcn_ds_swizzle(__float_as_int(x), 0x021F));
}

static __device__ __forceinline__ void store_v8(float* p, v8f v) {
  float4 a = {v[0], v[1], v[2], v[3]};
  float4 b = {v[4], v[5], v[6], v[7]};
  *(float4*)p = a; *(float4*)(p + 4) = b;
}
static __device__ __forceinline__ v8f load_v8(const float* p) {
  float4 a = *(const float4*)p;
  float4 b = *(const float4*)(p + 4);
  v8f v = {a.x, a.y, a.z, a.w, b.x, b.y, b.z, b.w};
  return v;
}

// Load one 16x4 f32 A-operand (A-layout: v0 = rows K0|K2, v1 = rows K1|K3 across
// lane halves; lane%16 = M = output column of the row-convention matrix M[k][m]).
static __device__ __forceinline__ v2f load_a(const float* Mrow, int ncols, int r0, int col0) {
  const int lane = threadIdx.x & 31;
  const int m = lane & 15;
  const int rr = r0 + ((lane < 16) ? 0 : 2);
  v2f a;
  a.x = Mrow[rr * ncols + col0 + m];
  a.y = Mrow[(rr + 1) * ncols + col0 + m];
  return a;
}

// Build the four 4x16 B-operand chunks (K rows 0-3,4-7,8-11,12-15) from a D-layout
// 16x16 tile: D.vg holds row g (lanes 0-15) and row g+8 (lanes 16-31).
static __device__ __forceinline__ void make_b_tile(v8f t, bool lo, v2f* bs) {
  float w0 = swz16(t[0]), w1 = swz16(t[1]), w2 = swz16(t[2]), w3 = swz16(t[3]);
  float w4 = swz16(t[4]), w5 = swz16(t[5]), w6 = swz16(t[6]), w7 = swz16(t[7]);
  bs[0].x = lo ? t[0] : w2;  bs[0].y = lo ? t[1] : w3;   // rows 0..3
  bs[1].x = lo ? t[4] : w6;  bs[1].y = lo ? t[5] : w7;   // rows 4..7
  bs[2].x = lo ? w0 : t[2];  bs[2].y = lo ? w1 : t[3];   // rows 8..11
  bs[3].x = lo ? w4 : t[6];  bs[3].y = lo ? w5 : t[7];   // rows 12..15
}
static __device__ __forceinline__ void make_bs(v8f s0, v8f s1, bool lo, v2f bs[8]) {
  make_b_tile(s0, lo, bs);
  make_b_tile(s1, lo, bs + 4);
}

struct StepConsts {
  v2f W[2][8];    // [out tile][K chunk]
  v2f W0[2][2];
  v2f W1[2][2];
};

static __device__ __forceinline__ void load_step_consts(const float* ws, StepConsts& c) {
#pragma unroll
  for (int n = 0; n < 2; ++n) {
#pragma unroll
    for (int k = 0; k < 8; ++k) c.W[n][k] = load_a(ws + W_OFF, NS, 4 * k, n * 16);
#pragma unroll
    for (int k = 0; k < 2; ++k) {
      c.W0[n][k] = load_a(ws + W0_OFF, NS, 4 * k, n * 16);
      c.W1[n][k] = load_a(ws + W1_OFF, NS, 4 * k, n * 16);
    }
  }
}

// One recurrence step: S' = W^T S + W0^T U0 + W1^T U1  (S held as two D tiles)
static __device__ __forceinline__ void state_step(const StepConsts& c, const v2f bs[8],
                                                  v2f u0k0, v2f u0k1, v2f u1k0, v2f u1k1,
                                                  v8f& s0, v8f& s1) {
  v8f n0 = {}, n1 = {};
  n0 = WMMA4(c.W0[0][0], u0k0, n0); n0 = WMMA4(c.W0[0][1], u0k1, n0);
  n0 = WMMA4(c.W1[0][0], u1k0, n0); n0 = WMMA4(c.W1[0][1], u1k1, n0);
  n1 = WMMA4(c.W0[1][0], u0k0, n1); n1 = WMMA4(c.W0[1][1], u0k1, n1);
  n1 = WMMA4(c.W1[1][0], u1k0, n1); n1 = WMMA4(c.W1[1][1], u1k1, n1);
#pragma unroll
  for (int k = 0; k < 8; ++k) {
    n0 = WMMA4(c.W[0][k], bs[k], n0);
    n1 = WMMA4(c.W[1][k], bs[k], n1);
  }
  s0 = n0; s1 = n1;
}

// ---------------------------------------------------------------------------
// Setup: derive the affine step operator from (A, B, dt) by basis propagation
// through one exact Dopri5 step, then W^L by repeated squaring.
// ---------------------------------------------------------------------------
static __device__ void rk_row(const float* A, const float* Bm, float dt,
                              float x[NS], const float u0[NI], const float u1[NI]) {
  float du[NI], us[NI];
  float k1[NS], k2[NS], k3[NS], k4[NS], k5[NS], k6[NS], xt[NS];
  for (int i = 0; i < NI; ++i) du[i] = u1[i] - u0[i];
  auto f = [&](const float* xx, const float* uu, float* out) {
    for (int s = 0; s < NS; ++s) {
      float a = 0.f;
      for (int k = 0; k < NS; ++k) a += A[s * NS + k] * xx[k];
      for (int i = 0; i < NI; ++i) a += Bm[s * NI + i] * uu[i];
      out[s] = a;
    }
  };
  f(x, u0, k1);
  for (int s = 0; s < NS; ++s) xt[s] = x[s] + dt * (0.2f * k1[s]);
  for (int i = 0; i < NI; ++i) us[i] = u0[i] + 0.2f * du[i];
  f(xt, us, k2);
  for (int s = 0; s < NS; ++s) xt[s] = x[s] + dt * ((3.f/40.f) * k1[s] + (9.f/40.f) * k2[s]);
  for (int i = 0; i < NI; ++i) us[i] = u0[i] + 0.3f * du[i];
  f(xt, us, k3);
  for (int s = 0; s < NS; ++s)
    xt[s] = x[s] + dt * ((44.f/45.f) * k1[s] - (56.f/15.f) * k2[s] + (32.f/9.f) * k3[s]);
  for (int i = 0; i < NI; ++i) us[i] = u0[i] + 0.8f * du[i];
  f(xt, us, k4);
  for (int s = 0; s < NS; ++s)
    xt[s] = x[s] + dt * ((19372.f/6561.f) * k1[s] - (25360.f/2187.f) * k2[s] +
                         (64448.f/6561.f) * k3[s] - (212.f/729.f) * k4[s]);
  for (int i = 0; i < NI; ++i) us[i] = u0[i] + (8.f/9.f) * du[i];
  f(xt, us, k5);
  for (int s = 0; s < NS; ++s)
    xt[s] = x[s] + dt * ((9017.f/3168.f) * k1[s] - (355.f/33.f) * k2[s] +
                         (46732.f/5247.f) * k3[s] + (49.f/176.f) * k4[s] -
                         (5103.f/18656.f) * k5[s]);
  f(xt, u1, k6);
  for (int s = 0; s < NS; ++s)
    x[s] = x[s] + dt * ((35.f/384.f) * k1[s] + (500.f/1113.f) * k3[s] +
                        (125.f/192.f) * k4[s] - (2187.f/6784.f) * k5[s] +
                        (11.f/84.f) * k6[s]);
}

__global__ void setup_kernel(const float* t, const float* A, const float* Bm,
                             const float* C, const float* D, float* ws) {
  const float dt = t[1] - t[0];
  const int tid = threadIdx.x;
  if (tid < 48) {
    float x[NS], u0[NI], u1[NI];
    for (int k = 0; k < NS; ++k) x[k] = 0.f;
    for (int i = 0; i < NI; ++i) { u0[i] = 0.f; u1[i] = 0.f; }
    float* dst;
    if (tid < 32)      { x[tid] = 1.f;        dst = ws + W_OFF  + tid * NS; }
    else if (tid < 40) { u0[tid - 32] = 1.f;  dst = ws + W0_OFF + (tid - 32) * NS; }
    else               { u1[tid - 40] = 1.f;  dst = ws + W1_OFF + (tid - 40) * NS; }
    rk_row(A, Bm, dt, x, u0, u1);
    for (int s = 0; s < NS; ++s) dst[s] = x[s];
  }
  for (int e = tid; e < NS * 16; e += blockDim.x) {
    int k = e / 16, o = e % 16;
    ws[CT_OFF + e] = (o < NO) ? C[o * NS + k] : 0.f;
  }
  for (int e = tid; e < NI * 16; e += blockDim.x) {
    int i = e / 16, o = e % 16;
    ws[DT_OFF + e] = (o < NO) ? D[o * NI + i] : 0.f;
  }
  __syncthreads();
  for (int e = tid; e < NS * NS; e += blockDim.x) ws[WL_OFF + e] = ws[W_OFF + e];
  __syncthreads();
  for (int r = 0; r < 6; ++r) {               // 2^6 = LCHUNK
    const float* src = ws + ((r & 1) ? TMP_OFF : WL_OFF);
    float* dst = ws + ((r & 1) ? WL_OFF : TMP_OFF);
    for (int e = tid; e < NS * NS; e += blockDim.x) {
      int i = e / NS, j = e % NS;
      float acc = 0.f;
      for (int k = 0; k < NS; ++k) acc += src[i * NS + k] * src[k * NS + j];
      dst[e] = acc;
    }
    __syncthreads();
  }
}

// ---------------------------------------------------------------------------
// Pass 1: per-chunk residual r_j = scan over chunk j starting from zero state.
// ---------------------------------------------------------------------------
__global__ void __launch_bounds__(32) pass1_kernel(const float* u, float* ws) {
  const int j = blockIdx.x;                 // 0 .. NCHUNK-2
  const int bt = blockIdx.y;
  const int lane = threadIdx.x & 31;
  const bool lo = lane < 16;
  const int b = bt * 16 + (lane & 15);
  StepConsts c; load_step_consts(ws, c);
  v8f s0 = {}, s1 = {};
  const int t0 = j * LCHUNK;
  const float* up = u + ((size_t)b * T_LEN + t0) * NI + (lo ? 0 : 2);
  float2 ca = *(const float2*)up;
  float2 cb = *(const float2*)(up + 4);
  v2f u0k0 = {ca.x, ca.y}, u0k1 = {cb.x, cb.y};
  for (int s = 0; s < LCHUNK; ++s) {
    up += NI;
    float2 na = *(const float2*)up;
    float2 nb = *(const float2*)(up + 4);
    v2f u1k0 = {na.x, na.y}, u1k1 = {nb.x, nb.y};
    v2f bs[8]; make_bs(s0, s1, lo, bs);
    state_step(c, bs, u0k0, u0k1, u1k0, u1k1, s0, s1);
    u0k0 = u1k0; u0k1 = u1k1;
  }
  float* rp = ws + R_OFF + (((size_t)(j * BT + bt) * 2) * 32 + lane) * 8;
  store_v8(rp, s0);
  store_v8(rp + 256, s1);
}

// ---------------------------------------------------------------------------
// Mid: sequential combine across chunks: x_{j+1} = x_j @ W^L + r_j.
// ---------------------------------------------------------------------------
__global__ void __launch_bounds__(32) mid_kernel(const float* x0, float* ws) {
  const int bt = blockIdx.x;
  const int lane = threadIdx.x & 31;
  const bool lo = lane < 16;
  const int b = bt * 16 + (lane & 15);
  v2f WLa[2][8];
#pragma unroll
  for (int n = 0; n < 2; ++n)
#pragma unroll
    for (int k = 0; k < 8; ++k) WLa[n][k] = load_a(ws + WL_OFF, NS, 4 * k, n * 16);
  const int sb = lo ? 0 : 8;
  v8f s0 = load_v8(x0 + b * NS + sb);
  v8f s1 = load_v8(x0 + b * NS + sb + 16);
  auto xoff = [&](int j, int tile) {
    return (((size_t)(j * BT + bt) * 2 + tile) * 32 + lane) * 8;
  };
  float* xsp = ws + XST_OFF;
  store_v8(xsp + xoff(0, 0), s0);
  store_v8(xsp + xoff(0, 1), s1);
  for (int j = 0; j < NCHUNK - 1; ++j) {
    v2f bs[8]; make_bs(s0, s1, lo, bs);
    v8f n0 = load_v8(ws + R_OFF + xoff(j, 0));
    v8f n1 = load_v8(ws + R_OFF + xoff(j, 1));
#pragma unroll
    for (int k = 0; k < 8; ++k) {
      n0 = WMMA4(WLa[0][k], bs[k], n0);
      n1 = WMMA4(WLa[1][k], bs[k], n1);
    }
    s0 = n0; s1 = n1;
    store_v8(xsp + xoff(j + 1, 0), s0);
    store_v8(xsp + xoff(j + 1, 1), s1);
  }
}

// ---------------------------------------------------------------------------
// Pass 3: re-scan each chunk from its true start state; emit xs and ys.
// ---------------------------------------------------------------------------
__global__ void __launch_bounds__(32) pass3_kernel(const float* u, const float* ws,
                                                   float* xs, float* ys) {
  const int j = blockIdx.x;
  const int bt = blockIdx.y;
  const int lane = threadIdx.x & 31;
  const bool lo = lane < 16;
  const int b = bt * 16 + (lane & 15);
  StepConsts c; load_step_consts(ws, c);
  v2f Ca[8], Da[2];
#pragma unroll
  for (int k = 0; k < 8; ++k) Ca[k] = load_a(ws + CT_OFF, 16, 4 * k, 0);
  Da[0] = load_a(ws + DT_OFF, 16, 0, 0);
  Da[1] = load_a(ws + DT_OFF, 16, 4, 0);

  const float* xsp = ws + XST_OFF;
  const size_t base = (((size_t)(j * BT + bt) * 2) * 32 + lane) * 8;
  v8f s0 = load_v8(xsp + base);
  v8f s1 = load_v8(xsp + base + 256);

  const int t0 = j * LCHUNK;
  const float* up = u + ((size_t)b * T_LEN + t0) * NI + (lo ? 0 : 2);
  float2 ca = *(const float2*)up;
  float2 cb = *(const float2*)(up + 4);
  v2f u0k0 = {ca.x, ca.y}, u0k1 = {cb.x, cb.y};

  float* xout = xs + ((size_t)b * T_LEN + t0) * NS + (lo ? 0 : 8);
  float* yout = ys + ((size_t)b * T_LEN + t0) * NO;

  for (int s = 0; s < LCHUNK; ++s) {
    v2f bs[8]; make_bs(s0, s1, lo, bs);
    // y_t = C^T x_t + D^T u_t (outputs in rows 0..7; rows 8..15 are zero-padded)
    v8f y = {};
    y = WMMA4(Da[0], u0k0, y);
    y = WMMA4(Da[1], u0k1, y);
#pragma unroll
    for (int k = 0; k < 8; ++k) y = WMMA4(Ca[k], bs[k], y);
    store_v8(xout, s0);
    store_v8(xout + 16, s1);
    if (lo) store_v8(yout, y);
    xout += NS; yout += NO;
    if (s < LCHUNK - 1) {
      up += NI;
      float2 na = *(const float2*)up;
      float2 nb = *(const float2*)(up + 4);
      v2f u1k0 = {na.x, na.y}, u1k1 = {nb.x, nb.y};
      state_step(c, bs, u0k0, u0k1, u1k0, u1k1, s0, s1);
      u0k0 = u1k0; u0k1 = u1k1;
    }
  }
}

extern "C" void kernel_launch(void* const* d_in, const int* in_sizes, int n_in,
                              void* d_out, int out_size, void* d_ws, size_t ws_size,
                              hipStream_t stream) {
  const float* t  = (const float*)d_in[0];
  const float* u  = (const float*)d_in[1];
  const float* x0 = (const float*)d_in[2];
  const float* A  = (const float*)d_in[3];
  const float* Bm = (const float*)d_in[4];
  const float* C  = (const float*)d_in[5];
  const float* D  = (const float*)d_in[6];
  float* ws = (float*)d_ws;
  float* xs = (float*)d_out;
  float* ys = xs + (size_t)BATCH * T_LEN * NS;

  setup_kernel<<<1, 256, 0, stream>>>(t, A, Bm, C, D, ws);
  pass1_kernel<<<dim3(NCHUNK - 1, BT), 32, 0, stream>>>(u, ws);
  mid_kernel<<<BT, 32, 0, stream>>>(x0, ws);
  pass3_kernel<<<dim3(NCHUNK, BT), 32, 0, stream>>>(u, ws, xs, ys);
}